// QNet_71296457113911
// MI455X (gfx1250) — compile-verified
//
#include <hip/hip_runtime.h>
#include <math.h>

// ---------------------------------------------------------------------------
// Problem constants (fixed by the reference)
// ---------------------------------------------------------------------------
static constexpr int   N1 = 100000, N2 = 50000, N3 = 25000;
static constexpr long  E1 = 1600000, E2 = 800000, E3 = 400000;
static constexpr int   A3LEN = 75000;
static constexpr int   HDIM = 64, NI2 = 32, NI3 = 64, G = 512, NCLS = 10;

typedef float v2f __attribute__((ext_vector_type(2)));
typedef float v8f __attribute__((ext_vector_type(8)));

__device__ __forceinline__ float elu1(float x) {
  return x > 0.f ? x : (expf(x) - 1.f);
}

// ---------------------------------------------------------------------------
// WMMA fp32 GEMM:  outR = act( rowscale .* (X@Wr) + bias ),  outN = X@Wn
// One wave computes a 16-row x (NB*16)-col tile with V_WMMA_F32_16X16X4_F32.
// K must be a multiple of 4. EXEC stays all-ones across all WMMA ops
// (row clamping for loads, predication only on the final stores).
// ---------------------------------------------------------------------------
template <int NB, bool DUAL>
__global__ void wmma_gemm_kernel(const float* __restrict__ X, int ldx, int K,
                                 const float* __restrict__ Wr,
                                 const float* __restrict__ Wn,
                                 const float* __restrict__ bias,
                                 const float* __restrict__ rowscale,
                                 float* __restrict__ outR,
                                 float* __restrict__ outN,
                                 int nrows, int act) {
  constexpr int Nout = NB * 16;
  const int lane = threadIdx.x & 31;
  const int wave = threadIdx.x >> 5;
  const int tile = blockIdx.x * (blockDim.x >> 5) + wave;
  const int row0 = tile * 16;
  if (row0 >= nrows) return;                 // wave-uniform exit

  const int m  = lane & 15;                  // M index for A, N index for B/C/D
  const int kh = lane >> 4;                  // which K pair {0,1} vs {2,3}

  v8f cR[NB] = {};
  v8f cN[NB] = {};

  int rowA = row0 + m;
  if (rowA >= nrows) rowA = nrows - 1;       // clamped load keeps EXEC full
  const float* xrow = X + (size_t)rowA * (size_t)ldx;

  for (int k = 0; k < K; k += 4) {
    v2f a;
    a.x = xrow[k + 2 * kh + 0];
    a.y = xrow[k + 2 * kh + 1];
    {
      const float* w0 = Wr + (size_t)(k + 2 * kh) * Nout + m;
      const float* w1 = w0 + Nout;
#pragma unroll
      for (int nb = 0; nb < NB; ++nb) {
        v2f b;
        b.x = w0[nb * 16];
        b.y = w1[nb * 16];
        cR[nb] = __builtin_amdgcn_wmma_f32_16x16x4_f32(
            false, a, false, b, (short)0, cR[nb], false, false);
      }
    }
    if (DUAL) {
      const float* w0 = Wn + (size_t)(k + 2 * kh) * Nout + m;
      const float* w1 = w0 + Nout;
#pragma unroll
      for (int nb = 0; nb < NB; ++nb) {
        v2f b;
        b.x = w0[nb * 16];
        b.y = w1[nb * 16];
        cN[nb] = __builtin_amdgcn_wmma_f32_16x16x4_f32(
            false, a, false, b, (short)0, cN[nb], false, false);
      }
    }
  }

  // C/D layout: VGPR r <-> row (row0 + r + 8*kh), col = nb*16 + m
#pragma unroll
  for (int nb = 0; nb < NB; ++nb) {
    const int col = nb * 16 + m;
    const float bv = bias ? bias[col] : 0.f;
#pragma unroll
    for (int r = 0; r < 8; ++r) {
      const int ro = row0 + r + 8 * kh;
      if (ro < nrows) {
        float v = cR[nb][r];
        if (rowscale) v *= rowscale[ro];
        v += bv;
        if (act) v = elu1(v);
        outR[(size_t)ro * Nout + col] = v;
        if (DUAL) outN[(size_t)ro * Nout + col] = cN[nb][r];
      }
    }
  }
}

// ---------------------------------------------------------------------------
// Edge message scatter:  out[dst] += hlin[src] * ec   (64 channels, fp32)
// 16 threads per edge, 4 channels each (float4 gather, 4x atomicAdd).
// Destination region is L2-resident (<= 25.6 MB), so atomics stay on-chip.
// ---------------------------------------------------------------------------
__global__ void edge_scatter_kernel(const float* __restrict__ hlin,
                                    const int* __restrict__ src,
                                    const int* __restrict__ dst,
                                    const float* __restrict__ ec,
                                    float* __restrict__ out, long E) {
  long t = (long)blockIdx.x * blockDim.x + threadIdx.x;
  long e = t >> 4;
  if (e >= E) return;
  int c4 = (int)(t & 15) * 4;
  int s = src[e], d = dst[e];
  float w = ec ? ec[e] : 1.0f;
  const float4 v = *(const float4*)(hlin + (size_t)s * HDIM + c4);
  float* o = out + (size_t)d * HDIM + c4;
  atomicAdd(o + 0, v.x * w);
  atomicAdd(o + 1, v.y * w);
  atomicAdd(o + 2, v.z * w);
  atomicAdd(o + 3, v.w * w);
}

// out[i] = elu(out[i])
__global__ void elu_kernel(float* __restrict__ x, long n) {
  long i = (long)blockIdx.x * blockDim.x + threadIdx.x;
  if (i < n) x[i] = elu1(x[i]);
}

__global__ void zero_kernel(float* __restrict__ x, long n) {
  long i = (long)blockIdx.x * blockDim.x + threadIdx.x;
  if (i < n) x[i] = 0.f;
}

// sums[seg[n]] += vals[n]; cnts[seg[n]] += 1 (optional). 16 threads/row.
__global__ void seg_accum_kernel(const float* __restrict__ vals,
                                 const int* __restrict__ seg,
                                 float* __restrict__ sums,
                                 float* __restrict__ cnts, long N) {
  long t = (long)blockIdx.x * blockDim.x + threadIdx.x;
  long n = t >> 4;
  if (n >= N) return;
  int c4 = (int)(t & 15) * 4;
  int g = seg[n];
  const float4 v = *(const float4*)(vals + (size_t)n * HDIM + c4);
  float* o = sums + (size_t)g * HDIM + c4;
  atomicAdd(o + 0, v.x);
  atomicAdd(o + 1, v.y);
  atomicAdd(o + 2, v.z);
  atomicAdd(o + 3, v.w);
  if (cnts && c4 == 0) atomicAdd(&cnts[g], 1.0f);
}

// sums[dst[i]] += h[src[i]]; cnts[dst[i]] += 1   (avg_pool accumulate)
__global__ void gather_seg_accum_kernel(const float* __restrict__ h,
                                        const int* __restrict__ src,
                                        const int* __restrict__ dst,
                                        float* __restrict__ sums,
                                        float* __restrict__ cnts, long M) {
  long t = (long)blockIdx.x * blockDim.x + threadIdx.x;
  long n = t >> 4;
  if (n >= M) return;
  int c4 = (int)(t & 15) * 4;
  int s = src[n], d = dst[n];
  const float4 v = *(const float4*)(h + (size_t)s * HDIM + c4);
  float* o = sums + (size_t)d * HDIM + c4;
  atomicAdd(o + 0, v.x);
  atomicAdd(o + 1, v.y);
  atomicAdd(o + 2, v.z);
  atomicAdd(o + 3, v.w);
  if (cnts && c4 == 0) atomicAdd(&cnts[d], 1.0f);
}

// out[n, 0:64] = sums[n]/max(cnt,1);  out[n, 64:64+DI] = iso[n]
__global__ void pool_finalize_kernel(const float* __restrict__ sums,
                                     const float* __restrict__ cnts,
                                     const float* __restrict__ iso,
                                     float* __restrict__ out, long N, int DI) {
  const int ldo = HDIM + DI;
  long t = (long)blockIdx.x * blockDim.x + threadIdx.x;
  long n = t / ldo;
  if (n >= N) return;
  int c = (int)(t % ldo);
  float v;
  if (c < HDIM)
    v = sums[(size_t)n * HDIM + c] / fmaxf(cnts[n], 1.0f);
  else
    v = iso[(size_t)n * DI + (c - HDIM)];
  out[(size_t)n * ldo + c] = v;
}

// z = concat(x1, s2/c2, s3/c3) -> (G, 192)
__global__ void pack_z_kernel(const float* __restrict__ x1,
                              const float* __restrict__ s2,
                              const float* __restrict__ c2,
                              const float* __restrict__ s3,
                              const float* __restrict__ c3,
                              float* __restrict__ z) {
  int t = blockIdx.x * blockDim.x + threadIdx.x;
  if (t >= G * 192) return;
  int g = t / 192, c = t % 192;
  float v;
  if (c < 64)       v = x1[g * 64 + c];
  else if (c < 128) v = s2[g * 64 + (c - 64)] / fmaxf(c2[g], 1.0f);
  else              v = s3[g * 64 + (c - 128)] / fmaxf(c3[g], 1.0f);
  z[t] = v;
}

// fc3 (32 -> 10) + log_softmax; one thread per graph row (512 rows).
__global__ void head_kernel(const float* __restrict__ f2,
                            const float* __restrict__ W,
                            const float* __restrict__ b,
                            float* __restrict__ out) {
  int g = blockIdx.x * blockDim.x + threadIdx.x;
  if (g >= G) return;
  float l[NCLS];
  for (int c = 0; c < NCLS; ++c) {
    float s = b[c];
    for (int k = 0; k < 32; ++k) s += f2[g * 32 + k] * W[k * NCLS + c];
    l[c] = s;
  }
  float mx = l[0];
  for (int c = 1; c < NCLS; ++c) mx = fmaxf(mx, l[c]);
  float sum = 0.f;
  for (int c = 0; c < NCLS; ++c) sum += expf(l[c] - mx);
  float lse = mx + logf(sum);
  for (int c = 0; c < NCLS; ++c) out[g * NCLS + c] = l[c] - lse;
}

// ---------------------------------------------------------------------------
// Host-side orchestration
// ---------------------------------------------------------------------------
static inline long cdivl(long a, long b) { return (a + b - 1) / b; }

extern "C" void kernel_launch(void* const* d_in, const int* in_sizes, int n_in,
                              void* d_out, int out_size, void* d_ws, size_t ws_size,
                              hipStream_t stream) {
  (void)in_sizes; (void)n_in; (void)out_size; (void)ws_size;

  // ---- inputs (setup_inputs dict order; params pytree flattens key-sorted:
  //      conv{1..7} each {Wn, Wr, b}, then fc{1,2,3} each {W, b}) ----
  const float* x      = (const float*)d_in[0];
  const int*   ei1    = (const int*)d_in[1];       // [0:E1]=src, [E1:2E1]=dst
  const float* nc     = (const float*)d_in[2];
  const float* ec     = (const float*)d_in[3];
  const int*   batch  = (const int*)d_in[4];
  const int*   a2s    = (const int*)d_in[5];
  const int*   a2d    = (const int*)d_in[6];
  const float* iso2   = (const float*)d_in[7];
  const int*   ei2    = (const int*)d_in[8];
  const int*   batch2 = (const int*)d_in[9];
  const int*   a3s    = (const int*)d_in[10];
  const int*   a3d    = (const int*)d_in[11];
  const float* iso3   = (const float*)d_in[12];
  const int*   ei3    = (const int*)d_in[13];
  const int*   batch3 = (const int*)d_in[14];
  // d_in[15] = num_graphs (compile-time constant G here)
  const float* cWn[7], *cWr[7], *cb[7];
  for (int i = 0; i < 7; ++i) {
    cWn[i] = (const float*)d_in[16 + 3 * i + 0];
    cWr[i] = (const float*)d_in[16 + 3 * i + 1];
    cb[i]  = (const float*)d_in[16 + 3 * i + 2];
  }
  const float* fc1W = (const float*)d_in[37];
  const float* fc1b = (const float*)d_in[38];
  const float* fc2W = (const float*)d_in[39];
  const float* fc2b = (const float*)d_in[40];
  const float* fc3W = (const float*)d_in[41];
  const float* fc3b = (const float*)d_in[42];
  float* out = (float*)d_out;

  // ---- workspace carve (fp32, 16B aligned) ----
  float* ws = (float*)d_ws;
  size_t off = 0;
  auto alloc = [&](size_t n) {
    float* p = ws + off;
    off += (n + 15) & ~(size_t)15;
    return p;
  };
  float* hA   = alloc((size_t)N1 * HDIM);
  float* hB   = alloc((size_t)N1 * HDIM);
  float* hT   = alloc((size_t)N1 * HDIM);
  float* x1   = alloc((size_t)G * HDIM);
  float* sum2 = alloc((size_t)N2 * HDIM);
  float* cnt2 = alloc((size_t)N2);
  float* h2in = alloc((size_t)N2 * (HDIM + NI2));
  float* h2a  = alloc((size_t)N2 * HDIM);
  float* h2t  = alloc((size_t)N2 * HDIM);
  float* s2   = alloc((size_t)G * HDIM);
  float* c2   = alloc((size_t)G);
  float* sum3 = alloc((size_t)N3 * HDIM);
  float* cnt3 = alloc((size_t)N3);
  float* h3in = alloc((size_t)N3 * (HDIM + NI3));
  float* h3a  = alloc((size_t)N3 * HDIM);
  float* h3t  = alloc((size_t)N3 * HDIM);
  float* s3   = alloc((size_t)G * HDIM);
  float* c3   = alloc((size_t)G);
  float* z    = alloc((size_t)G * 192);
  float* f1   = alloc((size_t)G * HDIM);
  float* f2   = alloc((size_t)G * 32);

  const int TPB = 256;
  auto zero = [&](float* p, long n) {
    zero_kernel<<<cdivl(n, TPB), TPB, 0, stream>>>(p, n);
  };

  // qconv: out = elu( rowscale.*(X@Wr) + bias + scatter_add(ec.*(X@Wn)[src] -> dst) )
  auto conv = [&](const float* Xin, int ldx, int K,
                  const float* Wn_, const float* Wr_, const float* b_,
                  const float* rowscale, const int* esrc, const int* edst,
                  const float* ew, long E, float* outBuf, float* tmpBuf, int nrows) {
    long tiles = cdivl(nrows, 16);
    long blocks = cdivl(tiles, TPB / 32);
    wmma_gemm_kernel<4, true><<<blocks, TPB, 0, stream>>>(
        Xin, ldx, K, Wr_, Wn_, b_, rowscale, outBuf, tmpBuf, nrows, /*act=*/0);
    edge_scatter_kernel<<<cdivl(E * 16, TPB), TPB, 0, stream>>>(
        tmpBuf, esrc, edst, ew, outBuf, E);
    long n = (long)nrows * HDIM;
    elu_kernel<<<cdivl(n, TPB), TPB, 0, stream>>>(outBuf, n);
  };

  // ---- level 1: three convs on (N1, 64) ----
  conv(x,  HDIM, HDIM, cWn[0], cWr[0], cb[0], nc, ei1, ei1 + E1, ec, E1, hA, hT, N1);
  conv(hA, HDIM, HDIM, cWn[1], cWr[1], cb[1], nc, ei1, ei1 + E1, ec, E1, hB, hT, N1);
  conv(hB, HDIM, HDIM, cWn[2], cWr[2], cb[2], nc, ei1, ei1 + E1, ec, E1, hA, hT, N1);

  // x1 = segment_sum(h, batch)
  zero(x1, (long)G * HDIM);
  seg_accum_kernel<<<cdivl((long)N1 * 16, TPB), TPB, 0, stream>>>(
      hA, batch, x1, nullptr, N1);

  // ---- level 2 ----
  zero(sum2, (long)N2 * HDIM);
  zero(cnt2, N2);
  gather_seg_accum_kernel<<<cdivl((long)N1 * 16, TPB), TPB, 0, stream>>>(
      hA, a2s, a2d, sum2, cnt2, N1);
  pool_finalize_kernel<<<cdivl((long)N2 * (HDIM + NI2), TPB), TPB, 0, stream>>>(
      sum2, cnt2, iso2, h2in, N2, NI2);
  conv(h2in, HDIM + NI2, HDIM + NI2, cWn[3], cWr[3], cb[3],
       nullptr, ei2, ei2 + E2, nullptr, E2, h2a, h2t, N2);
  conv(h2a, HDIM, HDIM, cWn[4], cWr[4], cb[4],
       nullptr, ei2, ei2 + E2, nullptr, E2, h2t, h2in /*tmp reuse*/, N2);
  zero(s2, (long)G * HDIM);
  zero(c2, G);
  seg_accum_kernel<<<cdivl((long)N2 * 16, TPB), TPB, 0, stream>>>(
      h2t, batch2, s2, c2, N2);

  // ---- level 3 ----
  zero(sum3, (long)N3 * HDIM);
  zero(cnt3, N3);
  gather_seg_accum_kernel<<<cdivl((long)A3LEN * 16, TPB), TPB, 0, stream>>>(
      hA, a3s, a3d, sum3, cnt3, A3LEN);
  pool_finalize_kernel<<<cdivl((long)N3 * (HDIM + NI3), TPB), TPB, 0, stream>>>(
      sum3, cnt3, iso3, h3in, N3, NI3);
  conv(h3in, HDIM + NI3, HDIM + NI3, cWn[5], cWr[5], cb[5],
       nullptr, ei3, ei3 + E3, nullptr, E3, h3a, h3t, N3);
  conv(h3a, HDIM, HDIM, cWn[6], cWr[6], cb[6],
       nullptr, ei3, ei3 + E3, nullptr, E3, h3t, h3in /*tmp reuse*/, N3);
  zero(s3, (long)G * HDIM);
  zero(c3, G);
  seg_accum_kernel<<<cdivl((long)N3 * 16, TPB), TPB, 0, stream>>>(
      h3t, batch3, s3, c3, N3);

  // ---- readout MLP ----
  pack_z_kernel<<<cdivl((long)G * 192, TPB), TPB, 0, stream>>>(x1, s2, c2, s3, c3, z);

  {
    long tiles = cdivl(G, 16), blocks = cdivl(tiles, TPB / 32);
    wmma_gemm_kernel<4, false><<<blocks, TPB, 0, stream>>>(
        z, 192, 192, fc1W, nullptr, fc1b, nullptr, f1, nullptr, G, /*act=*/1);
    wmma_gemm_kernel<2, false><<<blocks, TPB, 0, stream>>>(
        f1, HDIM, HDIM, fc2W, nullptr, fc2b, nullptr, f2, nullptr, G, /*act=*/1);
  }
  head_kernel<<<cdivl(G, TPB), TPB, 0, stream>>>(f2, fc3W, fc3b, out);
}